// MultiBoxLoss_4569845203248
// MI455X (gfx1250) — compile-verified
//
#include <hip/hip_runtime.h>
#include <hip/hip_bf16.h>

#define B_ 32
#define P_ 8732
#define M_ 16
#define C_ 81
#define THRESH 0.5f
#define RATIO 3

#define NT_B 256
#define SORT_N 16384
#define NT_S 1024
#define PAD_PART 1152  // >= 1092 blocks of kernel B, multiple of 64

typedef __attribute__((ext_vector_type(2))) float v2f;
typedef __attribute__((ext_vector_type(8))) float v8f;

// ---------------- Kernel 0: zero scratch accumulator regions ----------------
__global__ void MBL_zero_kernel(float* p, int n) {
    int i = blockIdx.x * 256 + threadIdx.x;
    if (i < n) p[i] = 0.0f;
}

// ---------------- Kernel A: matching (one block per image) ----------------
__global__ __launch_bounds__(256) void MBL_match_kernel(
    const float* __restrict__ boxes,    // [B,16,4] xy
    const int*   __restrict__ labels,   // [B,16]
    const float* __restrict__ priors,   // [P,4] cxcy
    float* __restrict__ true_locs,      // [B,P,4]
    int*   __restrict__ true_classes,   // [B,P]
    int*   __restrict__ n_pos)          // [B]
{
    __shared__ float sbox[M_][4];
    __shared__ float sarea[M_];
    __shared__ int   slab[M_];
    __shared__ float redv[M_][8];
    __shared__ int   redp[M_][8];
    __shared__ int   s_best_p[M_];
    __shared__ int   s_npos;

    const int b = blockIdx.x;
    const int tid = threadIdx.x;

    if (tid < M_ * 4) ((float*)sbox)[tid] = boxes[b * M_ * 4 + tid];
    if (tid < M_)     slab[tid] = labels[b * M_ + tid];
    if (tid == 0)     s_npos = 0;
    __syncthreads();
    if (tid < M_) sarea[tid] = (sbox[tid][2] - sbox[tid][0]) * (sbox[tid][3] - sbox[tid][1]);
    __syncthreads();

    // Pass 1: per-object best prior (per-thread, priors visited in ascending order)
    float bi[M_]; int bp[M_];
    #pragma unroll
    for (int m = 0; m < M_; ++m) { bi[m] = -1.0f; bp[m] = 0; }

    for (int p = tid; p < P_; p += 256) {
        float pcx = priors[p*4+0], pcy = priors[p*4+1];
        float pw  = priors[p*4+2], ph  = priors[p*4+3];
        float px0 = pcx - pw*0.5f, py0 = pcy - ph*0.5f;
        float px1 = pcx + pw*0.5f, py1 = pcy + ph*0.5f;
        float parea = (px1 - px0) * (py1 - py0);
        #pragma unroll
        for (int m = 0; m < M_; ++m) {
            float lx = fmaxf(sbox[m][0], px0), ly = fmaxf(sbox[m][1], py0);
            float rx = fminf(sbox[m][2], px1), ry = fminf(sbox[m][3], py1);
            float w = fmaxf(rx - lx, 0.0f), h = fmaxf(ry - ly, 0.0f);
            float inter = w * h;
            float iou = inter / (sarea[m] + parea - inter);
            if (iou > bi[m]) { bi[m] = iou; bp[m] = p; }  // strict >: keeps lowest p
        }
    }

    // wave32 butterfly reduce, tie-break: smaller prior index wins (matches argmax first-occurrence)
    #pragma unroll
    for (int m = 0; m < M_; ++m) {
        float v = bi[m]; int p = bp[m];
        #pragma unroll
        for (int off = 16; off > 0; off >>= 1) {
            float v2 = __shfl_xor(v, off, 32);
            int   p2 = __shfl_xor(p, off, 32);
            if (v2 > v || (v2 == v && p2 < p)) { v = v2; p = p2; }
        }
        if ((tid & 31) == 0) { redv[m][tid >> 5] = v; redp[m][tid >> 5] = p; }
    }
    __syncthreads();
    if (tid < M_) {
        float best = -2.0f; int bpv = 0;
        for (int w = 0; w < 8; ++w) {
            float v = redv[tid][w]; int p = redp[tid][w];
            if (v > best || (v == best && p < bpv)) { best = v; bpv = p; }
        }
        s_best_p[tid] = bpv;
    }
    __syncthreads();

    // Pass 2: per-prior best object (recomputed), forced matches, labels + encoded locs
    int my_pos = 0;
    for (int p = tid; p < P_; p += 256) {
        float pcx = priors[p*4+0], pcy = priors[p*4+1];
        float pw  = priors[p*4+2], ph  = priors[p*4+3];
        float px0 = pcx - pw*0.5f, py0 = pcy - ph*0.5f;
        float px1 = pcx + pw*0.5f, py1 = pcy + ph*0.5f;
        float parea = (px1 - px0) * (py1 - py0);
        float best = -1.0f; int obj = 0;
        #pragma unroll
        for (int m = 0; m < M_; ++m) {
            float lx = fmaxf(sbox[m][0], px0), ly = fmaxf(sbox[m][1], py0);
            float rx = fminf(sbox[m][2], px1), ry = fminf(sbox[m][3], py1);
            float w = fmaxf(rx - lx, 0.0f), h = fmaxf(ry - ly, 0.0f);
            float inter = w * h;
            float iou = inter / (sarea[m] + parea - inter);
            if (iou > best) { best = iou; obj = m; }  // first max wins
        }
        float ov = best;
        #pragma unroll
        for (int m = 0; m < M_; ++m)
            if (s_best_p[m] == p) { obj = m; ov = 1.0f; }  // sequential scatter: last m wins

        int lab = slab[obj];
        if (ov < THRESH) lab = 0;
        true_classes[b * P_ + p] = lab;
        if (lab != 0) my_pos++;

        float x0 = sbox[obj][0], y0 = sbox[obj][1], x1 = sbox[obj][2], y1 = sbox[obj][3];
        float cx = 0.5f * (x0 + x1), cy = 0.5f * (y0 + y1);
        float w = x1 - x0, h = y1 - y0;
        float* tl = true_locs + ((size_t)(b * P_ + p)) * 4;
        tl[0] = (cx - pcx) / (pw * 0.1f);
        tl[1] = (cy - pcy) / (ph * 0.1f);
        tl[2] = logf(w / pw) * 5.0f;
        tl[3] = logf(h / ph) * 5.0f;
    }
    atomicAdd(&s_npos, my_pos);
    __syncthreads();
    if (tid == 0) n_pos[b] = s_npos;
}

// ---------------- Kernel B: per-prior CE + loc L1 (bandwidth-bound, 90.5MB) ----------------
__global__ __launch_bounds__(NT_B) void MBL_conf_loc_kernel(
    const float* __restrict__ plocs,    // [B,P,4]
    const float* __restrict__ plogits,  // [B,P,81]
    const float* __restrict__ tlocs,    // [B,P,4]
    const int*   __restrict__ tcls,     // [B,P]
    float* __restrict__ conf_neg,       // [B,P]
    float* __restrict__ part_conf,      // [PAD_PART]
    float* __restrict__ part_loc)       // [PAD_PART]
{
    const int idx = blockIdx.x * NT_B + threadIdx.x;
    float conf_pos = 0.0f, loc = 0.0f;
    if (idx < B_ * P_) {
        const float* row = plogits + (size_t)idx * C_;
        __builtin_prefetch(row, 0, 0);            // global_prefetch_b8
        __builtin_prefetch(row + 64, 0, 0);
        float mx = row[0];
        for (int c = 1; c < C_; ++c) mx = fmaxf(mx, row[c]);
        float s = 0.0f;
        for (int c = 0; c < C_; ++c) s += expf(row[c] - mx);
        const int lab = tcls[idx];
        float cl = (mx + logf(s)) - row[lab];     // -log_softmax[lab]
        if (!isfinite(cl)) cl = 0.0f;             // nan_to_num
        if (lab != 0) {
            conf_pos = cl;
            conf_neg[idx] = 0.0f;
            const float* pl = plocs + (size_t)idx * 4;
            const float* tl = tlocs + (size_t)idx * 4;
            loc = fabsf(pl[0]-tl[0]) + fabsf(pl[1]-tl[1]) +
                  fabsf(pl[2]-tl[2]) + fabsf(pl[3]-tl[3]);
        } else {
            conf_neg[idx] = cl;
        }
    }
    __shared__ float r1[NT_B], r2[NT_B];
    r1[threadIdx.x] = conf_pos; r2[threadIdx.x] = loc;
    __syncthreads();
    for (int s2 = NT_B / 2; s2 > 0; s2 >>= 1) {
        if (threadIdx.x < s2) {
            r1[threadIdx.x] += r1[threadIdx.x + s2];
            r2[threadIdx.x] += r2[threadIdx.x + s2];
        }
        __syncthreads();
    }
    if (threadIdx.x == 0) { part_conf[blockIdx.x] = r1[0]; part_loc[blockIdx.x] = r2[0]; }
}

// ---------------- Kernel C: hard-negative mining via in-LDS bitonic sort ----------------
// 16384 floats = 64KB LDS (CDNA5 WGP has 320KB). Top-K sum is tie-order invariant.
__global__ __launch_bounds__(NT_S) void MBL_hardneg_kernel(
    const float* __restrict__ conf_neg, const int* __restrict__ n_pos,
    float* __restrict__ neg_sum)
{
    __shared__ float s[SORT_N];
    const int b = blockIdx.x, tid = threadIdx.x;
    for (int i = tid; i < SORT_N; i += NT_S)
        s[i] = (i < P_) ? conf_neg[b * P_ + i] : -1.0f;   // real CE >= 0, pad sinks to bottom
    __syncthreads();
    for (int k = 2; k <= SORT_N; k <<= 1) {
        for (int j = k >> 1; j > 0; j >>= 1) {
            for (int i = tid; i < SORT_N; i += NT_S) {
                int ixj = i ^ j;
                if (ixj > i) {
                    bool up = ((i & k) == 0);            // ascending overall
                    float a = s[i], c = s[ixj];
                    if (up ? (a > c) : (a < c)) { s[i] = c; s[ixj] = a; }
                }
            }
            __syncthreads();
        }
    }
    int K = RATIO * n_pos[b];
    if (K > P_) K = P_;
    float acc = 0.0f;
    for (int i = tid; i < K; i += NT_S) acc += s[SORT_N - 1 - i];
    __syncthreads();
    s[tid] = acc;                                         // reuse buffer
    __syncthreads();
    for (int s2 = NT_S / 2; s2 > 0; s2 >>= 1) {
        if (tid < s2) s[tid] += s[tid + s2];
        __syncthreads();
    }
    if (tid == 0) neg_sum[b] = s[0];
}

// ---------------- Kernel D: final reduction via WMMA ones-matmul (single wave) --------
// D = A(16x4) x ones(4x16) + C accumulates exact f32 row sums: 64 values per v_wmma.
__global__ __launch_bounds__(32) void MBL_final_kernel(
    const float* __restrict__ part_conf,  // [PAD_PART], zero padded
    const float* __restrict__ part_loc,   // [PAD_PART], zero padded
    const float* __restrict__ neg_sum,    // [64], zero padded
    const int*   __restrict__ n_pos,      // [B]
    float* __restrict__ out)
{
    const int lane = threadIdx.x;
    const int mrow = lane & 15;
    const int koff = (lane >> 4) * 2;     // ISA A-layout: lanes 0-15 hold K=0,1; 16-31 hold K=2,3
    v8f acc_c = {}, acc_l = {}, acc_n = {};
    v2f ones; ones.x = 1.0f; ones.y = 1.0f;   // all-ones B: layout-independent

    for (int ch = 0; ch < PAD_PART; ch += 64) {
        const int base = ch + mrow * 4 + koff;
        v2f a; a.x = part_conf[base]; a.y = part_conf[base + 1];
        acc_c = __builtin_amdgcn_wmma_f32_16x16x4_f32(false, a, false, ones, (short)0, acc_c, false, false);
        v2f l; l.x = part_loc[base]; l.y = part_loc[base + 1];
        acc_l = __builtin_amdgcn_wmma_f32_16x16x4_f32(false, l, false, ones, (short)0, acc_l, false, false);
    }
    {
        const int base = mrow * 4 + koff;
        v2f a; a.x = neg_sum[base]; a.y = neg_sum[base + 1];
        acc_n = __builtin_amdgcn_wmma_f32_16x16x4_f32(false, a, false, ones, (short)0, acc_n, false, false);
    }

    // D layout: vgpr r, lane 0 -> (M=r, N=0); lane 16 -> (M=8+r, N=0). Sum rows 0..15.
    __shared__ float sred[6];
    float t_c = 0.0f, t_l = 0.0f, t_n = 0.0f;
    #pragma unroll
    for (int r = 0; r < 8; ++r) { t_c += acc_c[r]; t_l += acc_l[r]; t_n += acc_n[r]; }
    if (lane == 0)  { sred[0] = t_c; sred[1] = t_l; sred[2] = t_n; }
    if (lane == 16) { sred[3] = t_c; sred[4] = t_l; sred[5] = t_n; }
    __syncthreads();
    if (lane == 0) {
        float conf_pos = sred[0] + sred[3];
        float loc      = sred[1] + sred[4];
        float neg      = sred[2] + sred[5];
        float npt = 0.0f;
        for (int i = 0; i < B_; ++i) npt += (float)n_pos[i];
        float loc_loss  = loc / (4.0f * npt);
        float conf_loss = (neg + conf_pos) / npt;
        out[0] = conf_loss + loc_loss;   // ALPHA = 1
    }
}

// ---------------- launcher ----------------
extern "C" void kernel_launch(void* const* d_in, const int* in_sizes, int n_in,
                              void* d_out, int out_size, void* d_ws, size_t ws_size,
                              hipStream_t stream) {
    const float* plocs   = (const float*)d_in[0];   // [B,P,4]
    const float* plogits = (const float*)d_in[1];   // [B,P,81]
    const float* boxes   = (const float*)d_in[2];   // [B,16,4]
    const int*   labels  = (const int*)  d_in[3];   // [B,16]
    const float* priors  = (const float*)d_in[4];   // [P,4]
    float* out = (float*)d_out;

    float* ws = (float*)d_ws;
    size_t o = 0;
    float* true_locs = ws + o;      o += (size_t)B_ * P_ * 4;
    int*   tcls      = (int*)(ws + o); o += (size_t)B_ * P_;
    float* cneg      = ws + o;      o += (size_t)B_ * P_;
    int*   npos      = (int*)(ws + o); o += 64;
    float* pconf     = ws + o;      o += PAD_PART;
    float* ploc      = ws + o;      o += PAD_PART;
    float* nsum      = ws + o;      o += 64;
    // total ~6.7MB of d_ws

    const int zn = 2 * PAD_PART + 64;   // pconf|ploc|nsum are contiguous
    MBL_zero_kernel<<<(zn + 255) / 256, 256, 0, stream>>>(pconf, zn);

    MBL_match_kernel<<<B_, 256, 0, stream>>>(boxes, labels, priors, true_locs, tcls, npos);

    const int nblk = (B_ * P_ + NT_B - 1) / NT_B;   // 1092 <= PAD_PART
    MBL_conf_loc_kernel<<<nblk, NT_B, 0, stream>>>(plocs, plogits, true_locs, tcls,
                                                   cneg, pconf, ploc);

    MBL_hardneg_kernel<<<B_, NT_S, 0, stream>>>(cneg, npos, nsum);

    MBL_final_kernel<<<1, 32, 0, stream>>>(pconf, ploc, nsum, npos, out);
}